// GCNLayer_89807766159585
// MI455X (gfx1250) — compile-verified
//
#include <hip/hip_runtime.h>

// ---------------------------------------------------------------------------
// GCN layer on MI455X (gfx1250):  out = relu(BN((A@X)@W) + bias)
// A: [N,N] f32 (1.6 GB => the whole cost), X: [N,32], W: [32,32]
// A@X has AI = 16 FLOP/byte ~= fp32-WMMA ridge (~310 TF / 23.3 TB/s ~ 13),
// so the main loop must be pure {load A frag, 2x WMMA} with A read once.
// ---------------------------------------------------------------------------

#define NN   20000
#define NIN  32
#define NOUT 32
#define KSPLIT 5
#define KCHUNK (NN / KSPLIT)        // 4000, multiple of 32

typedef __attribute__((ext_vector_type(2))) float v2f;
typedef __attribute__((ext_vector_type(8))) float v8f;

// ws layout (floats)
#define WS_AX   0                         // [NN*32]  A@X accumulator
#define WS_XT   (NN * 32)                 // [32*NN]  X transposed
#define WS_H    (2 * NN * 32)             // [NN*32]  h = AX@W
#define WS_SUM  (3 * NN * 32)             // [32]     column sums of h
#define WS_SSQ  (3 * NN * 32 + 32)        // [32]     column sums of h^2

// ---------------------------------------------------------------------------
// Kernel 0: zero AX + stats, build XT[c][r] = X[r][c]
// ---------------------------------------------------------------------------
__global__ void k0_init(const float* __restrict__ X, float* __restrict__ ws) {
    int tid = blockIdx.x * blockDim.x + threadIdx.x;
    if (tid < NN * 32) {
        ws[WS_AX + tid] = 0.0f;
        int r = tid >> 5, c = tid & 31;
        ws[WS_XT + (size_t)c * NN + r] = X[tid];
    }
    if (tid < 64) ws[WS_SUM + tid] = 0.0f;   // SUM and SSQ are contiguous
}

// ---------------------------------------------------------------------------
// Kernel 1: AX += A[:, k0:k0+KCHUNK] @ X[k0:k0+KCHUNK, :]  via fp32 WMMA.
// Wave -> 16 rows x ALL 32 cols (two 16x16 accumulators), so each A-fragment
// load feeds two V_WMMA_F32_16X16X4_F32. Block = 5 waves = 80 rows.
// grid = (250, KSPLIT). Fragment layouts per CDNA5 ISA 7.12.2:
//   A 16x4 : lane L holds row m0+(L&15); VGPR pair = K {koff, koff+1},
//            koff = (L<16) ? 0 : 2  -> one aligned float2 load from A.
//   B 4x16 : lane L holds col n0+(L&15); same K pairing -> float2 from XT.
// One global_prefetch_b8 per 128B A-line (outer step = 32 K-elements).
// ---------------------------------------------------------------------------
__global__ void k1_wmma_ax(const float* __restrict__ A,
                           const float* __restrict__ ws_xt,
                           float* __restrict__ ws_ax) {
    const unsigned lane = threadIdx.x & 31u;
    const unsigned wave = threadIdx.x >> 5;          // 0..4
    const int m0 = blockIdx.x * 80 + (int)wave * 16;
    const int k0 = blockIdx.y * KCHUNK;
    const int koff = (lane < 16u) ? 0 : 2;
    const int half = (int)(lane & 15u);

    const float* __restrict__ aptr  = A     + (size_t)(m0 + half) * NN + (k0 + koff);
    const float* __restrict__ bptr0 = ws_xt + (size_t)(half)      * NN + (k0 + koff);
    const float* __restrict__ bptr1 = ws_xt + (size_t)(16 + half) * NN + (k0 + koff);

    v8f acc0 = {};
    v8f acc1 = {};
    for (int kk = 0; kk < KCHUNK; kk += 32) {
        // one cacheline of A per row, prefetched 8KB ahead of the compute front
        __builtin_prefetch(aptr + kk + 2048, 0, 1);
#pragma unroll
        for (int t = 0; t < 32; t += 4) {
            v2f a  = *(const v2f*)(aptr  + kk + t);
            v2f b0 = *(const v2f*)(bptr0 + kk + t);
            v2f b1 = *(const v2f*)(bptr1 + kk + t);
            acc0 = __builtin_amdgcn_wmma_f32_16x16x4_f32(
                false, a, false, b0, (short)0, acc0, false, false);
            acc1 = __builtin_amdgcn_wmma_f32_16x16x4_f32(
                false, a, false, b1, (short)0, acc1, false, false);
        }
    }

    // D layout: VGPR r, lanes 0-15 -> M=r, lanes 16-31 -> M=r+8; N = lane&15
    const int mbase = m0 + ((lane < 16u) ? 0 : 8);
#pragma unroll
    for (int r = 0; r < 8; ++r) {
        float* rowp = ws_ax + (size_t)(mbase + r) * NOUT;
        __hip_atomic_fetch_add(rowp + half, acc0[r],
                               __ATOMIC_RELAXED, __HIP_MEMORY_SCOPE_AGENT);
        __hip_atomic_fetch_add(rowp + 16 + half, acc1[r],
                               __ATOMIC_RELAXED, __HIP_MEMORY_SCOPE_AGENT);
    }
}

// ---------------------------------------------------------------------------
// Kernel 2: h = AX @ W  (K=32), plus per-column sum / sum^2 for BatchNorm.
// block (32,8): 8 rows x 32 cols; W staged in LDS; LDS fp32 atomics reduce.
// ---------------------------------------------------------------------------
__global__ void k2_transform(const float* __restrict__ ws_ax,
                             const float* __restrict__ W,
                             float* __restrict__ ws_h,
                             float* __restrict__ gsum,
                             float* __restrict__ gssq) {
    __shared__ float sW[NIN * NOUT];
    __shared__ float ssum[NOUT];
    __shared__ float sssq[NOUT];

    const int col = threadIdx.x;                      // 0..31
    const int row = blockIdx.x * 8 + threadIdx.y;     // 0..NN-1
    const int tid = threadIdx.y * 32 + threadIdx.x;   // 0..255

#pragma unroll
    for (int i = tid; i < NIN * NOUT; i += 256) sW[i] = W[i];
    if (tid < NOUT) { ssum[tid] = 0.0f; sssq[tid] = 0.0f; }
    __syncthreads();

    const float* __restrict__ axr = ws_ax + (size_t)row * NIN;
    float acc = 0.0f;
#pragma unroll
    for (int i = 0; i < NIN; ++i) acc = fmaf(axr[i], sW[i * NOUT + col], acc);

    ws_h[(size_t)row * NOUT + col] = acc;
    atomicAdd(&ssum[col], acc);            // ds_add_f32
    atomicAdd(&sssq[col], acc * acc);
    __syncthreads();

    if (tid < NOUT) {
        __hip_atomic_fetch_add(&gsum[tid], ssum[tid],
                               __ATOMIC_RELAXED, __HIP_MEMORY_SCOPE_AGENT);
        __hip_atomic_fetch_add(&gssq[tid], sssq[tid],
                               __ATOMIC_RELAXED, __HIP_MEMORY_SCOPE_AGENT);
    }
}

// ---------------------------------------------------------------------------
// Kernel 3: BN finalize + bias + ReLU -> d_out (fp32)
// ---------------------------------------------------------------------------
__global__ void k3_finalize(const float* __restrict__ ws_h,
                            const float* __restrict__ gsum,
                            const float* __restrict__ gssq,
                            const float* __restrict__ gamma,
                            const float* __restrict__ beta,
                            const float* __restrict__ bias,
                            float* __restrict__ out) {
    int tid = blockIdx.x * blockDim.x + threadIdx.x;
    if (tid >= NN * NOUT) return;
    const int col = tid & 31;
    const float inv_n = 1.0f / (float)NN;
    const float mean = gsum[col] * inv_n;
    const float var  = gssq[col] * inv_n - mean * mean;
    const float rs   = rsqrtf(var + 1e-5f);
    float v = (ws_h[tid] - mean) * rs * gamma[col] + beta[col] + bias[col];
    out[tid] = v > 0.0f ? v : 0.0f;
}

// ---------------------------------------------------------------------------
extern "C" void kernel_launch(void* const* d_in, const int* in_sizes, int n_in,
                              void* d_out, int out_size, void* d_ws, size_t ws_size,
                              hipStream_t stream) {
    const float* A     = (const float*)d_in[0];
    const float* X     = (const float*)d_in[1];
    const float* W     = (const float*)d_in[2];
    const float* gamma = (const float*)d_in[3];
    const float* beta  = (const float*)d_in[4];
    const float* bias  = (const float*)d_in[5];
    float* out = (float*)d_out;
    float* ws  = (float*)d_ws;

    // 0: zero AX/stats, build XT                 (2500 blocks x 256)
    k0_init<<<(NN * 32 + 255) / 256, 256, 0, stream>>>(X, ws);

    // 1: WMMA main GEMM, split-K atomics          (250 x 5 blocks x 160)
    k1_wmma_ax<<<dim3(NN / 80, KSPLIT), 160, 0, stream>>>(A, ws + WS_XT, ws + WS_AX);

    // 2: h = AX@W + BN partial sums               (2500 blocks x (32,8))
    k2_transform<<<NN / 8, dim3(32, 8), 0, stream>>>(ws + WS_AX, W, ws + WS_H,
                                                     ws + WS_SUM, ws + WS_SSQ);

    // 3: BN + bias + ReLU                         (2500 blocks x 256)
    k3_finalize<<<(NN * NOUT + 255) / 256, 256, 0, stream>>>(ws + WS_H, ws + WS_SUM,
                                                             ws + WS_SSQ, gamma, beta,
                                                             bias, out);
}